// GatedMultiAggHead_1898375545099
// MI455X (gfx1250) — compile-verified
//
#include <hip/hip_runtime.h>
#include <math.h>

#define HDIM 128
#define BSEG 512
#define LN_EPS 1e-5f

typedef float v4f __attribute__((ext_vector_type(4)));
typedef float v2f __attribute__((ext_vector_type(2)));
typedef float v8f __attribute__((ext_vector_type(8)));

// ---------------------------------------------------------------------------
// lower_bound over sorted int32 segment ids
// ---------------------------------------------------------------------------
__device__ __forceinline__ int lower_bound_i32(const int* __restrict__ b, int n, int key) {
    int lo = 0, hi = n;
    while (lo < hi) {
        int mid = (lo + hi) >> 1;
        if (b[mid] < key) lo = mid + 1; else hi = mid;
    }
    return lo;
}

__device__ __forceinline__ float sigmoidf_fast(float x) {
    return 1.0f / (1.0f + __expf(-x));
}

// ---------------------------------------------------------------------------
// Kernel 1: gated multi-aggregation pooling for one rank.
//   grid.x = BSEG (one workgroup per segment), 256 threads = 8 wave32
//   agg[s, 0:128]=sum, [128:256]=mean, [256:384]=max(guarded), [384:512]=gsum
// Streaming-bound: NT float4 loads, butterfly shuffle for the gate dot.
// ---------------------------------------------------------------------------
__global__ __launch_bounds__(256) void pool_kernel(
    const float* __restrict__ h, const int* __restrict__ bseg, int N,
    const float* __restrict__ Wg, const float* __restrict__ bg,
    float* __restrict__ agg /* [BSEG, 4*HDIM] */) {

    const int s = blockIdx.x;
    __shared__ int bounds[2];
    if (threadIdx.x == 0) bounds[0] = lower_bound_i32(bseg, N, s);
    if (threadIdx.x == 1) bounds[1] = lower_bound_i32(bseg, N, s + 1);
    __syncthreads();
    const int start = bounds[0];
    const int end   = bounds[1];
    const int cnt   = end - start;

    const int lane = threadIdx.x & 31;
    const int wv   = threadIdx.x >> 5;   // 0..7
    const int nw   = blockDim.x >> 5;    // 8

    // gate weights: lane owns 4 contiguous columns
    const v4f wg  = *(const v4f*)(Wg + lane * 4);
    const float bgv = bg[0];

    v4f sum  = {0.f, 0.f, 0.f, 0.f};
    v4f gsum = {0.f, 0.f, 0.f, 0.f};
    v4f mx   = {-INFINITY, -INFINITY, -INFINITY, -INFINITY};

    for (int i = start + wv; i < end; i += nw) {
        const float* rowp = h + (size_t)i * HDIM + lane * 4;
        // prefetch this wave's next row (speculative; OOB is safe)
        __builtin_prefetch(rowp + (size_t)nw * HDIM, 0, 0);
        // streaming read: non-temporal 128-bit load
        const v4f v = __builtin_nontemporal_load((const v4f*)rowp);

        // gate: dot(row, Wg) via lane partial + wave32 butterfly
        float p = v.x * wg.x + v.y * wg.y + v.z * wg.z + v.w * wg.w;
        #pragma unroll
        for (int off = 16; off >= 1; off >>= 1)
            p += __shfl_xor(p, off, 32);
        const float g = sigmoidf_fast(p + bgv);

        sum.x += v.x; sum.y += v.y; sum.z += v.z; sum.w += v.w;
        mx.x = fmaxf(mx.x, v.x); mx.y = fmaxf(mx.y, v.y);
        mx.z = fmaxf(mx.z, v.z); mx.w = fmaxf(mx.w, v.w);
        gsum.x += g * v.x; gsum.y += g * v.y;
        gsum.z += g * v.z; gsum.w += g * v.w;
    }

    // cross-wave combine through LDS
    __shared__ float red[3][8][HDIM];
    *(v4f*)&red[0][wv][lane * 4] = sum;
    *(v4f*)&red[1][wv][lane * 4] = mx;
    *(v4f*)&red[2][wv][lane * 4] = gsum;
    __syncthreads();

    if (threadIdx.x < HDIM) {
        const int j = threadIdx.x;
        float ssum = 0.f, smax = -INFINITY, sg = 0.f;
        #pragma unroll
        for (int w = 0; w < 8; ++w) {
            ssum += red[0][w][j];
            smax = fmaxf(smax, red[1][w][j]);
            sg   += red[2][w][j];
        }
        const float mean = ssum / fmaxf((float)cnt, 1.0f);
        if (cnt == 0) smax = 0.f;   // reference guards empty segments
        float* out = agg + (size_t)s * (4 * HDIM);
        out[j]              = ssum;
        out[HDIM + j]       = mean;
        out[2 * HDIM + j]   = smax;
        out[3 * HDIM + j]   = sg;
    }
}

// ---------------------------------------------------------------------------
// Kernel 2: r_r = agg_r[512,512] @ Wp_r[512,128] + bp_r via V_WMMA_F32_16X16X4_F32
//   grid = (Mtiles=32, Ntiles=8, rank=3), 32 threads (one wave, EXEC all ones)
//   Writes into state[512, 384] at column offset rank*128.
// Fragment layouts per ISA 7.12.2 (32-bit A 16x4; B mirrored; C/D 8 VGPRs).
// ---------------------------------------------------------------------------
__global__ __launch_bounds__(32) void gemm_wmma_kernel(
    const float* __restrict__ aggbase,  // 3 * 512 * 512
    const float* __restrict__ Wp0, const float* __restrict__ Wp1, const float* __restrict__ Wp2,
    const float* __restrict__ bp0, const float* __restrict__ bp1, const float* __restrict__ bp2,
    float* __restrict__ state /* [BSEG, 3*HDIM] */) {

    const int mt = blockIdx.x;      // 0..31  (rows of 16)
    const int nt = blockIdx.y;      // 0..7   (cols of 16)
    const int r  = blockIdx.z;      // rank

    const float* __restrict__ A  = aggbase + (size_t)r * BSEG * (4 * HDIM);
    const float* __restrict__ W  = (r == 0) ? Wp0 : ((r == 1) ? Wp1 : Wp2);
    const float* __restrict__ bp = (r == 0) ? bp0 : ((r == 1) ? bp1 : bp2);

    const int lane = threadIdx.x;
    const int half = lane >> 4;         // 0/1 selects K pair
    const int l15  = lane & 15;
    const int arow = mt * 16 + l15;     // A: M = lane&15
    const int bcol = nt * 16 + l15;     // B/D: N = lane&15

    v8f acc = {0.f, 0.f, 0.f, 0.f, 0.f, 0.f, 0.f, 0.f};

    #pragma unroll 4
    for (int k = 0; k < 4 * HDIM; k += 4) {
        const int ka = k + half * 2;
        v2f a, b;
        a.x = A[(size_t)arow * (4 * HDIM) + ka];
        a.y = A[(size_t)arow * (4 * HDIM) + ka + 1];
        b.x = W[(size_t)ka * HDIM + bcol];
        b.y = W[(size_t)(ka + 1) * HDIM + bcol];
        acc = __builtin_amdgcn_wmma_f32_16x16x4_f32(
            /*neg_a=*/false, a, /*neg_b=*/false, b,
            /*c_mod=*/(short)0, acc, /*reuse_a=*/false, /*reuse_b=*/false);
    }

    const float bias = bp[bcol];
    #pragma unroll
    for (int v = 0; v < 8; ++v) {
        const int row = mt * 16 + v + 8 * half;   // D: M = vgpr + 8*(lane>>4)
        state[(size_t)row * (3 * HDIM) + r * HDIM + bcol] = acc[v] + bias;
    }
}

// ---------------------------------------------------------------------------
// Kernel 3: LayerNorm + SiLU + [384->128] + SiLU + [128->1]
//   grid = BSEG rows, 128 threads (4 wave32)
// ---------------------------------------------------------------------------
__global__ __launch_bounds__(128) void head_kernel(
    const float* __restrict__ state,   // [BSEG, 384]
    const float* __restrict__ gamma, const float* __restrict__ beta,
    const float* __restrict__ W1, const float* __restrict__ b1f,  // [384,128],[128]
    const float* __restrict__ W2, const float* __restrict__ b2f,  // [128],[1]
    float* __restrict__ out /* [BSEG] */) {

    const int row  = blockIdx.x;
    const int tid  = threadIdx.x;
    const int lane = tid & 31;
    const int wv   = tid >> 5;    // 0..3

    __shared__ float xs[3 * HDIM];
    __shared__ float rs[4], rq[4], rr[4];

    const float* st = state + (size_t)row * (3 * HDIM);
    const float a0 = st[tid];
    const float a1 = st[tid + HDIM];
    const float a2 = st[tid + 2 * HDIM];

    float s = a0 + a1 + a2;
    float q = a0 * a0 + a1 * a1 + a2 * a2;
    #pragma unroll
    for (int off = 16; off >= 1; off >>= 1) {
        s += __shfl_xor(s, off, 32);
        q += __shfl_xor(q, off, 32);
    }
    if (lane == 0) { rs[wv] = s; rq[wv] = q; }
    __syncthreads();
    const float S  = rs[0] + rs[1] + rs[2] + rs[3];
    const float Q  = rq[0] + rq[1] + rq[2] + rq[3];
    const float mu = S * (1.0f / (3 * HDIM));
    const float var = Q * (1.0f / (3 * HDIM)) - mu * mu;
    const float rstd = rsqrtf(var + LN_EPS);

    // LayerNorm + SiLU, 3 columns per thread
    {
        const float x0 = (a0 - mu) * rstd * gamma[tid] + beta[tid];
        const float x1 = (a1 - mu) * rstd * gamma[tid + HDIM] + beta[tid + HDIM];
        const float x2 = (a2 - mu) * rstd * gamma[tid + 2 * HDIM] + beta[tid + 2 * HDIM];
        xs[tid]            = x0 * sigmoidf_fast(x0);
        xs[tid + HDIM]     = x1 * sigmoidf_fast(x1);
        xs[tid + 2 * HDIM] = x2 * sigmoidf_fast(x2);
    }
    __syncthreads();

    // x1[tid] = silu( dot(xs, W1[:,tid]) + b1f[tid] )
    float acc = b1f[tid];
    #pragma unroll 4
    for (int k = 0; k < 3 * HDIM; ++k)
        acc = fmaf(xs[k], W1[(size_t)k * HDIM + tid], acc);
    const float x1v = acc * sigmoidf_fast(acc);

    // out[row] = dot(x1, W2) + b2f
    float prod = x1v * W2[tid];
    #pragma unroll
    for (int off = 16; off >= 1; off >>= 1)
        prod += __shfl_xor(prod, off, 32);
    if (lane == 0) rr[wv] = prod;
    __syncthreads();
    if (tid == 0)
        out[row] = rr[0] + rr[1] + rr[2] + rr[3] + b2f[0];
}

// ---------------------------------------------------------------------------
// Host launcher
// ---------------------------------------------------------------------------
extern "C" void kernel_launch(void* const* d_in, const int* in_sizes, int n_in,
                              void* d_out, int out_size, void* d_ws, size_t ws_size,
                              hipStream_t stream) {
    (void)n_in; (void)out_size; (void)ws_size;

    const float* h0  = (const float*)d_in[0];
    const float* h1  = (const float*)d_in[1];
    const float* h2  = (const float*)d_in[2];
    const int*   b0  = (const int*)d_in[3];
    const int*   b1  = (const int*)d_in[4];
    const int*   b2  = (const int*)d_in[5];
    const float* Wg0 = (const float*)d_in[6];
    const float* bg0 = (const float*)d_in[7];
    const float* Wg1 = (const float*)d_in[8];
    const float* bg1 = (const float*)d_in[9];
    const float* Wg2 = (const float*)d_in[10];
    const float* bg2 = (const float*)d_in[11];
    const float* Wp0 = (const float*)d_in[12];
    const float* bp0 = (const float*)d_in[13];
    const float* Wp1 = (const float*)d_in[14];
    const float* bp1 = (const float*)d_in[15];
    const float* Wp2 = (const float*)d_in[16];
    const float* bp2 = (const float*)d_in[17];
    const float* gamma = (const float*)d_in[18];
    const float* beta  = (const float*)d_in[19];
    const float* W1  = (const float*)d_in[20];
    const float* b1f = (const float*)d_in[21];
    const float* W2  = (const float*)d_in[22];
    const float* b2f = (const float*)d_in[23];
    float* out = (float*)d_out;

    const int N0 = in_sizes[0] / HDIM;
    const int N1 = in_sizes[1] / HDIM;
    const int N2 = in_sizes[2] / HDIM;

    // workspace: agg[3][512][512] then state[512][384]
    float* agg   = (float*)d_ws;
    float* state = agg + (size_t)3 * BSEG * (4 * HDIM);

    pool_kernel<<<BSEG, 256, 0, stream>>>(h0, b0, N0, Wg0, bg0, agg + (size_t)0 * BSEG * 4 * HDIM);
    pool_kernel<<<BSEG, 256, 0, stream>>>(h1, b1, N1, Wg1, bg1, agg + (size_t)1 * BSEG * 4 * HDIM);
    pool_kernel<<<BSEG, 256, 0, stream>>>(h2, b2, N2, Wg2, bg2, agg + (size_t)2 * BSEG * 4 * HDIM);

    gemm_wmma_kernel<<<dim3(BSEG / 16, HDIM / 16, 3), 32, 0, stream>>>(
        agg, Wp0, Wp1, Wp2, bp0, bp1, bp2, state);

    head_kernel<<<BSEG, HDIM, 0, stream>>>(state, gamma, beta, W1, b1f, W2, b2f, out);
}